// _GATConvGlobalBondLayer_56951266345488
// MI455X (gfx1250) — compile-verified
//
#include <hip/hip_runtime.h>

// ---------------------------------------------------------------------------
// GATConvGlobalBondLayer forward for MI455X (gfx1250, wave32, WMMA).
//
// Pipeline (all on `stream`):
//   1) h = atom_feats   @ W_atom^T   -> d_ws           (f32 WMMA GEMM)
//   2) u = global_feats @ W_global^T -> d_ws           (f32 WMMA GEMM)
//   3) e = bond_feats   @ W_bond^T   -> d_out          (f32 WMMA GEMM)
//   4) e += h[src] + h[dst] + u[atom_mol[src]]; per-block partial sum/sumsq
//   5) deterministic reduction of partials -> sums[256]
//   6) BatchNorm(batch stats) + ReLU in place on d_out
//
// h (154MB) + u (5MB) fit in the 192MB L2, so the bond-gathers in step 4
// should be L2-resident; total HBM traffic ~1GB -> memory-bound design.
// ---------------------------------------------------------------------------

typedef __attribute__((ext_vector_type(2))) float v2f;
typedef __attribute__((ext_vector_type(8))) float v8f;

#define DIM 128
#define BN_EPS 1e-5f

// OUT[N][128] = X[N][128] * W[128][128]^T  (out[n][d] = sum_k X[n][k]*W[d][k])
// Block = 256 threads = 8 waves; each wave owns a 16-row x 128-col strip.
// Wave fragment mapping for V_WMMA_F32_16X16X4_F32 (ISA 7.12.2):
//   A vgpr j, lane l : A[l%16][2*(l/16)+j]
//   B vgpr j, lane l : B[2*(l/16)+j][l%16]   with B[k][n] = W[n][k]
//   D vgpr r, lane l : D[r+8*(l/16)][l%16]
__global__ __launch_bounds__(256) void gemm_xwt(const float* __restrict__ X,
                                                const float* __restrict__ W,
                                                float* __restrict__ OUT,
                                                int N) {
  // K-sliced W tile: 128 rows x 32 cols, +4 pad -> conflict-free ds_load_b64
  __shared__ float wlds[128][36];

  const int tid  = threadIdx.x;
  const int wave = tid >> 5;
  const int lane = tid & 31;
  const int half = lane >> 4;   // 0: lanes 0-15, 1: lanes 16-31
  const int lmod = lane & 15;
  const int m_base = blockIdx.x * 128 + wave * 16;

  // Clamp A row: an out-of-range A row only feeds out-of-range D rows,
  // and those stores are guarded below. Keeps EXEC uniform around WMMA.
  int arow = m_base + lmod;
  if (arow > N - 1) arow = N - 1;
  const float* __restrict__ xrow = X + (size_t)arow * DIM;

  v8f acc[8] = {};

  for (int kc = 0; kc < DIM; kc += 32) {
    __syncthreads();
#pragma unroll
    for (int i = tid; i < 128 * 32; i += 256) {
      const int r = i >> 5;
      const int c = i & 31;
      wlds[r][c] = W[(size_t)r * DIM + kc + c];
    }
    __syncthreads();

#pragma unroll
    for (int k0 = 0; k0 < 32; k0 += 4) {
      const int kk = k0 + 2 * half;
      v2f a;
      a.x = xrow[kc + kk];
      a.y = xrow[kc + kk + 1];
#pragma unroll
      for (int nt = 0; nt < 8; ++nt) {
        v2f b;
        b.x = wlds[nt * 16 + lmod][kk];
        b.y = wlds[nt * 16 + lmod][kk + 1];
        acc[nt] = __builtin_amdgcn_wmma_f32_16x16x4_f32(
            /*neg_a=*/false, a, /*neg_b=*/false, b,
            /*c_mod=*/(short)0, acc[nt], /*reuse_a=*/false, /*reuse_b=*/false);
      }
    }
  }

#pragma unroll
  for (int nt = 0; nt < 8; ++nt) {
#pragma unroll
    for (int r = 0; r < 8; ++r) {
      const int orow = m_base + r + 8 * half;
      if (orow < N) OUT[(size_t)orow * DIM + nt * 16 + lmod] = acc[nt][r];
    }
  }
}

// e[row] += h[src] + h[dst] + u[atom_mol[src]]; emit per-block partial
// sum / sumsq per feature column (deterministic two-stage reduction).
// Block handles 64 rows: thread (c = tid%128, half = tid/128) walks rows
// rowBase + half + 2*i  -> two coalesced 512B row segments per iteration.
__global__ __launch_bounds__(256) void assemble_stats(
    const float* __restrict__ h, const float* __restrict__ u,
    const int* __restrict__ bsrc, const int* __restrict__ bdst,
    const int* __restrict__ amol, float* __restrict__ e,
    float* __restrict__ partials, int n_bond) {
  __shared__ float red[512];
  const int tid = threadIdx.x;
  const int c   = tid & 127;
  const int hlf = tid >> 7;
  const long rowBase = (long)blockIdx.x * 64;

  float s = 0.0f, sq = 0.0f;
  for (int i = 0; i < 32; ++i) {
    const long row = rowBase + hlf + 2 * i;
    if (row < n_bond) {
      const int is = bsrc[row];
      const int id = bdst[row];
      const int ig = amol[is];
      const size_t off = (size_t)row * DIM + c;
      float v = e[off] + h[(size_t)is * DIM + c] + h[(size_t)id * DIM + c] +
                u[(size_t)ig * DIM + c];
      e[off] = v;
      s += v;
      sq += v * v;
    }
  }
  red[tid] = s;
  red[256 + tid] = sq;
  __syncthreads();
  if (tid < 128) {
    partials[(size_t)blockIdx.x * 256 + tid] = red[tid] + red[tid + 128];
    partials[(size_t)blockIdx.x * 256 + 128 + tid] =
        red[256 + tid] + red[384 + tid];
  }
}

// Fixed-order reduction of per-block partials -> sums[0:128]=sum,
// sums[128:256]=sumsq. Single block, fully deterministic.
__global__ __launch_bounds__(256) void reduce_partials(
    const float* __restrict__ partials, float* __restrict__ sums,
    int nblocks) {
  const int tid = threadIdx.x;
  float a = 0.0f;
  for (int b = 0; b < nblocks; ++b) a += partials[(size_t)b * 256 + tid];
  sums[tid] = a;
}

// In-place BatchNorm (biased batch variance) + affine + ReLU.
__global__ __launch_bounds__(256) void bn_relu(
    float* __restrict__ e, const float* __restrict__ sums,
    const float* __restrict__ gamma, const float* __restrict__ beta,
    int n_bond) {
  const int tid = threadIdx.x;
  const int c   = tid & 127;
  const int hlf = tid >> 7;
  const float inv_n = 1.0f / (float)n_bond;
  const float mean = sums[c] * inv_n;
  float var = sums[128 + c] * inv_n - mean * mean;
  var = var < 0.0f ? 0.0f : var;
  const float scale = gamma[c] * rsqrtf(var + BN_EPS);
  const float shift = beta[c] - mean * scale;
  const long rowBase = (long)blockIdx.x * 64;
  for (int i = 0; i < 32; ++i) {
    const long row = rowBase + hlf + 2 * i;
    if (row < n_bond) {
      const size_t off = (size_t)row * DIM + c;
      const float v = e[off] * scale + shift;
      e[off] = v > 0.0f ? v : 0.0f;
    }
  }
}

extern "C" void kernel_launch(void* const* d_in, const int* in_sizes, int n_in,
                              void* d_out, int out_size, void* d_ws,
                              size_t ws_size, hipStream_t stream) {
  (void)n_in; (void)out_size; (void)ws_size;

  const float* atom   = (const float*)d_in[0];
  const float* bond   = (const float*)d_in[1];
  const float* glob   = (const float*)d_in[2];
  const int*   bsrc   = (const int*)d_in[3];
  const int*   bdst   = (const int*)d_in[4];
  const int*   amol   = (const int*)d_in[5];
  const float* W_atom = (const float*)d_in[6];
  const float* W_bond = (const float*)d_in[7];
  const float* W_glob = (const float*)d_in[8];
  const float* gamma  = (const float*)d_in[9];
  const float* beta   = (const float*)d_in[10];

  const int n_atom = in_sizes[0] / DIM;
  const int n_bond = in_sizes[1] / DIM;
  const int n_glob = in_sizes[2] / DIM;

  // d_out doubles as the e buffer (written by GEMM, updated in place twice).
  float* e = (float*)d_out;

  // Workspace layout: h | u | sums(256) | partials
  float* h        = (float*)d_ws;
  float* u        = h + (size_t)n_atom * DIM;
  float* sums     = u + (size_t)n_glob * DIM;
  float* partials = sums + 256;

  const int gb_atom = (n_atom + 127) / 128;
  const int gb_bond = (n_bond + 127) / 128;
  const int gb_glob = (n_glob + 127) / 128;
  const int rb      = (n_bond + 63) / 64;

  gemm_xwt<<<gb_atom, 256, 0, stream>>>(atom, W_atom, h, n_atom);
  gemm_xwt<<<gb_glob, 256, 0, stream>>>(glob, W_glob, u, n_glob);
  gemm_xwt<<<gb_bond, 256, 0, stream>>>(bond, W_bond, e, n_bond);
  assemble_stats<<<rb, 256, 0, stream>>>(h, u, bsrc, bdst, amol, e, partials,
                                         n_bond);
  reduce_partials<<<1, 256, 0, stream>>>(partials, sums, rb);
  bn_relu<<<rb, 256, 0, stream>>>(e, sums, gamma, beta, n_bond);
}